// adder_layer_5617817223690
// MI455X (gfx1250) — compile-verified
//
#include <hip/hip_runtime.h>
#include <stdint.h>

// AdderNet forward, MI455X (gfx1250, wave32).
// X[8,32,32,32] NHWC fp32, filters[64,3,3,32] fp32.
// out[n,h,w,f] = sum_{dh,dw,c} | Xpad[n, h-1+dh, w-1+dw, c] - F[f,dh,dw,c] |

#define N_DIM   8
#define H_DIM   32
#define W_DIM   32
#define C_DIM   32
#define F_DIM   64
#define K_TOT   288          // 3*3*32, ordered (dh, dw, c) to match reference
#define FPB     32           // filters per block (two blocks cover all 64)
#define FPW     4            // filters per wave (8 waves * 4 = 32)
#define XW      36           // padded c-stride for sx rows: 36 = 4 mod 8 words
                             // -> only 2-way LDS bank conflict on b128 reads

__global__ __launch_bounds__(256)
void addernet_l1_fwd(const float* __restrict__ X,
                     const float* __restrict__ F,
                     float* __restrict__ out)
{
    __shared__ __align__(16) float sfilt[FPB * K_TOT];   // 36864 B, layout [f][288]
    __shared__ __align__(16) float sx[3 * 34 * XW];      // 14688 B, [dh][w+1][c(pad 36)]

    const int tid   = threadIdx.x;
    const int bx    = blockIdx.x;        // n*32 + h
    const int fhalf = blockIdx.y;        // 0..1
    const int n     = bx >> 5;
    const int h     = bx & 31;
    const int wave  = tid >> 5;          // wave32 on gfx1250
    const int lane  = tid & 31;          // = output w coordinate (coalesced)
    const int fbase = wave * FPW;        // filter index inside this block's slice

    const unsigned sfilt_lds = (unsigned)(uint64_t)(void*)&sfilt[0];
    const unsigned sx_lds    = (unsigned)(uint64_t)(void*)&sx[0];

    // ---- async copy: this block's 32 filters -> LDS (2304 x b128, 9/thread) ----
    {
        const unsigned gbase = (unsigned)fhalf * (unsigned)(FPB * K_TOT * 4);
        #pragma unroll
        for (int i = 0; i < (FPB * K_TOT / 4) / 256; ++i) {
            unsigned idx  = (unsigned)tid + (unsigned)i * 256u;
            unsigned goff = gbase + idx * 16u;
            unsigned loff = sfilt_lds + idx * 16u;
            asm volatile("global_load_async_to_lds_b128 %0, %1, %2"
                         :: "v"(loff), "v"(goff), "s"(F) : "memory");
        }
    }

    // ---- halo columns (wsrc = -1 and wsrc = 32) <- 0 (|0 - f| still counts) ----
    if (tid < 192) {
        int dh = tid / 64;
        int wi = ((tid & 63) >= 32) ? 33 : 0;
        int c  = tid & 31;
        sx[(dh * 34 + wi) * XW + c] = 0.0f;
    }

    // ---- async copy: input rows h-1..h+1 -> LDS (256 x b128/plane, 1/thread) ----
    #pragma unroll
    for (int dh = 0; dh < 3; ++dh) {
        int hsrc = h - 1 + dh;
        if (hsrc >= 0 && hsrc < H_DIM) {            // block-uniform branch
            unsigned wsrc = (unsigned)tid >> 3;               // 0..31
            unsigned c0   = ((unsigned)tid & 7u) * 4u;        // 0,4,...,28
            unsigned goff = ((((unsigned)(n * H_DIM + hsrc) * W_DIM) + wsrc) * C_DIM + c0) * 4u;
            unsigned loff = sx_lds + (((unsigned)(dh * 34) + wsrc + 1u) * XW + c0) * 4u;
            asm volatile("global_load_async_to_lds_b128 %0, %1, %2"
                         :: "v"(loff), "v"(goff), "s"(X) : "memory");
        } else {
            // whole plane is zero padding
            for (int idx = tid; idx < 34 * 32; idx += 256) {
                int wi = idx >> 5, c = idx & 31;
                sx[(dh * 34 + wi) * XW + c] = 0.0f;
            }
        }
    }

#if __has_builtin(__builtin_amdgcn_s_wait_asynccnt)
    __builtin_amdgcn_s_wait_asynccnt(0);
#else
    asm volatile("s_wait_asynccnt 0" ::: "memory");
#endif
    __syncthreads();

    // ---- L1 accumulation: pure VALU floor (v_sub_f32 + v_add_f32 acc, acc, |d|)
    float acc[FPW] = {0.f, 0.f, 0.f, 0.f};

    for (int dh = 0; dh < 3; ++dh) {
        for (int dw = 0; dw < 3; ++dw) {
            // wsrc = lane-1+dw  ->  halo index = lane+dw  (0..33)
            const float* xrow = &sx[(dh * 34 + lane + dw) * XW];
            // filter k index = (dh*3+dw)*32 + c ; reads are wave-uniform -> broadcast
            const float* frow = &sfilt[fbase * K_TOT + (dh * 3 + dw) * C_DIM];
            #pragma unroll
            for (int c4 = 0; c4 < C_DIM; c4 += 4) {
                const float4 xv = *(const float4*)&xrow[c4];
                #pragma unroll
                for (int ff = 0; ff < FPW; ++ff) {
                    const float4 fv = *(const float4*)&frow[ff * K_TOT + c4];
                    acc[ff] += __builtin_fabsf(xv.x - fv.x);
                    acc[ff] += __builtin_fabsf(xv.y - fv.y);
                    acc[ff] += __builtin_fabsf(xv.z - fv.z);
                    acc[ff] += __builtin_fabsf(xv.w - fv.w);
                }
            }
        }
    }

    // ---- store out[((n*32+h)*32 + w)*64 + fhalf*32 + fbase ...], one b128 ----
    float* orow = out + ((size_t)bx * W_DIM + lane) * F_DIM + fhalf * FPB + fbase;
    *(float4*)orow = make_float4(acc[0], acc[1], acc[2], acc[3]);
}

extern "C" void kernel_launch(void* const* d_in, const int* in_sizes, int n_in,
                              void* d_out, int out_size, void* d_ws, size_t ws_size,
                              hipStream_t stream)
{
    (void)in_sizes; (void)n_in; (void)out_size; (void)d_ws; (void)ws_size;
    const float* X  = (const float*)d_in[0];
    const float* Fw = (const float*)d_in[1];
    float* out      = (float*)d_out;

    dim3 grid(N_DIM * H_DIM, 2);   // 256 (n,h) rows x 2 filter halves = 512 blocks
    dim3 block(256);               // 8 wave32 waves
    addernet_l1_fwd<<<grid, block, 0, stream>>>(X, Fw, out);
}